// DynamicMicroscaleAttention_12472585027941
// MI455X (gfx1250) — compile-verified
//
#include <hip/hip_runtime.h>
#include <hip/hip_bf16.h>

// ---------------- problem constants (fixed by the reference) ----------------
#define B_   8
#define L_   2048
#define H_   8
#define HD_  64
#define DM_  512          // H_*HD_
#define WIN_ 6
#define STEP_ 3
#define NW_  681          // (L_-OV)//STEP
#define LN_EPS 1e-5f

typedef float v2f __attribute__((ext_vector_type(2)));
typedef float v8f __attribute__((ext_vector_type(8)));

// D = A(16x4,f32) * B(4x16,f32) + C(16x16,f32)
__device__ __forceinline__ v8f wmma4(v2f a, v2f b, v8f c) {
    return __builtin_amdgcn_wmma_f32_16x16x4_f32(
        /*neg_a=*/false, a, /*neg_b=*/false, b,
        /*c_mod=*/(short)0, c, /*reuse_a=*/false, /*reuse_b=*/false);
}

// ============================================================================
// Kernel 1: local windowed attention (replicates the reference's scrambled
// reshape).  One block (128 thr = 4 waves) per group g=(b,h2,n2).
// Each wave owns a 16-wide e-tile of the 16x64 projection outputs.
// ============================================================================
__global__ void local_attn_kernel(
    const float* __restrict__ q, const float* __restrict__ k,
    const float* __restrict__ v,
    const float* __restrict__ Wq, const float* __restrict__ Wk,
    const float* __restrict__ Wv,
    const float* __restrict__ temp_p,
    float* __restrict__ acc, float* __restrict__ aw_out)
{
    int g  = blockIdx.x;                 // 0 .. B*H*NW-1
    int n2 = g % NW_;
    int r0 = g / NW_;
    int h2 = r0 % H_;
    int b  = r0 / H_;

    __shared__ float qs[WIN_][HD_];
    __shared__ float ks_[WIN_][HD_];
    __shared__ float vs[WIN_][HD_];
    __shared__ float sc[WIN_][WIN_];
    __shared__ float pw[WIN_][WIN_];

    int tid  = threadIdx.x;
    int lane = tid & 31;
    int wv   = tid >> 5;
    int half = lane >> 4;
    int khalf = half << 1;
    int nlane = lane & 15;

    // per-lane A-fragment source row (rows 6..15 are padding -> alias row 0)
    int rj = (nlane < WIN_) ? nlane : 0;
    int f  = h2 * (NW_ * WIN_) + n2 * WIN_ + rj;   // scrambled flat index
    int n  = f / (WIN_ * H_);
    int rem = f % (WIN_ * H_);
    int w  = rem / H_;
    int hh = rem % H_;
    int tok = n * STEP_ + w;
    size_t base = ((size_t)(b * L_ + tok) * H_ + hh) * HD_;
    const float* xq = q + base;
    const float* xk = k + base;
    const float* xv = v + base;

    int e0 = wv * 16;
    v8f cq = {}, ck = {}, cv = {};
    #pragma unroll
    for (int kb = 0; kb < HD_; kb += 4) {
        int kk = kb + khalf;
        v2f aq; aq.x = xq[kk]; aq.y = xq[kk + 1];
        v2f ak; ak.x = xk[kk]; ak.y = xk[kk + 1];
        v2f av; av.x = xv[kk]; av.y = xv[kk + 1];
        int e = e0 + nlane;
        v2f bq = *(const v2f*)(Wq + (size_t)e * HD_ + kk);
        v2f bk = *(const v2f*)(Wk + (size_t)e * HD_ + kk);
        v2f bv = *(const v2f*)(Wv + (size_t)e * HD_ + kk);
        cq = wmma4(aq, bq, cq);
        ck = wmma4(ak, bk, ck);
        cv = wmma4(av, bv, cv);
    }
    // store valid rows (0..5) to LDS; those live in half==0, r<6
    if (half == 0) {
        #pragma unroll
        for (int r = 0; r < 8; ++r) {
            if (r < WIN_) {
                qs[r][e0 + nlane]  = cq[r];
                ks_[r][e0 + nlane] = ck[r];
                vs[r][e0 + nlane]  = cv[r];
            }
        }
    }
    __syncthreads();

    float invt = 1.0f / temp_p[0];
    if (tid < WIN_ * WIN_) {
        int i = tid / WIN_, j = tid % WIN_;
        float s = 0.f;
        #pragma unroll 16
        for (int d = 0; d < HD_; ++d) s += qs[i][d] * ks_[j][d];
        sc[i][j] = s * invt;
    }
    __syncthreads();

    if (tid < WIN_) {
        int i = tid;
        float m = -3.0e38f;
        #pragma unroll
        for (int j = 0; j < WIN_; ++j) m = fmaxf(m, sc[i][j]);
        float ssum = 0.f;
        float e[WIN_];
        #pragma unroll
        for (int j = 0; j < WIN_; ++j) { e[j] = __expf(sc[i][j] - m); ssum += e[j]; }
        float rs = 1.0f / ssum;
        size_t awb = ((size_t)((b * H_ + h2) * NW_ + n2)) * (WIN_ * WIN_) + (size_t)i * WIN_;
        #pragma unroll
        for (int j = 0; j < WIN_; ++j) {
            float p = e[j] * rs;
            pw[i][j] = p;
            aw_out[awb + j] = p;
        }
    }
    __syncthreads();

    // ao = P @ V, scatter-add into acc[b, n2*3+i, h2*64+d]  (lw == 1)
    for (int idx = tid; idx < WIN_ * HD_; idx += blockDim.x) {
        int i = idx / HD_, d = idx % HD_;
        float s = 0.f;
        #pragma unroll
        for (int j = 0; j < WIN_; ++j) s += pw[i][j] * vs[j][d];
        int tt = n2 * STEP_ + i;
        atomicAdd(&acc[((size_t)(b * L_ + tt)) * DM_ + h2 * HD_ + d], s);
    }
}

// ============================================================================
// Kernel 2: global projections  Y = X @ W^T   (X: L x 512, W: 512 x 512)
// grid (L/16, B, 3); block 128 = 4 waves; A tile staged in LDS (32 KB).
// ============================================================================
__global__ void proj_g_kernel(
    const float* __restrict__ q, const float* __restrict__ k,
    const float* __restrict__ v,
    const float* __restrict__ Wq, const float* __restrict__ Wk,
    const float* __restrict__ Wv,
    float* __restrict__ Qg, float* __restrict__ Kg, float* __restrict__ Vg)
{
    int mt = blockIdx.x, b = blockIdx.y, which = blockIdx.z;
    const float* X; const float* W; float* Y;
    if (which == 0)      { X = q; W = Wq; Y = Qg; }
    else if (which == 1) { X = k; W = Wk; Y = Kg; }
    else                 { X = v; W = Wv; Y = Vg; }
    X += (size_t)b * L_ * DM_;
    Y += (size_t)b * L_ * DM_;

    __shared__ float xa[16][DM_];          // 32 KB
    int tid  = threadIdx.x;
    {
        const float4* src = (const float4*)(X + (size_t)mt * 16 * DM_);
        float4* dst = (float4*)(&xa[0][0]);
        for (int i = tid; i < 16 * DM_ / 4; i += 128) dst[i] = src[i];
    }
    __syncthreads();

    int lane = tid & 31;
    int wv   = tid >> 5;
    int half = lane >> 4;
    int khalf = half << 1;
    int nlane = lane & 15;

    for (int nt = wv; nt < DM_ / 16; nt += 4) {
        int e0 = nt * 16;
        const float* wrow = W + (size_t)(e0 + nlane) * DM_;
        v8f c = {};
        #pragma unroll 8
        for (int kb = 0; kb < DM_; kb += 4) {
            int kk = kb + khalf;
            v2f a  = *(const v2f*)&xa[nlane][kk];       // ds_load_b64
            v2f bb = *(const v2f*)(wrow + kk);          // W^T fragment
            c = wmma4(a, bb, c);
        }
        float* yp = Y + (size_t)mt * 16 * DM_;
        #pragma unroll
        for (int r = 0; r < 8; ++r)
            yp[(size_t)(r + half * 8) * DM_ + e0 + nlane] = c[r];
    }
}

// ============================================================================
// Kernel 3: global attention, flash-style online softmax.
// grid (L/16, B); block 128 = 4 waves; wave wv owns d-chunk [wv*128, +128)
// for both the S partial (reduced across waves via LDS f32 atomics) and O/V.
// ============================================================================
__global__ void global_attn_kernel(
    const float* __restrict__ Qg, const float* __restrict__ Kg,
    const float* __restrict__ Vg, const float* __restrict__ temp_p,
    float* __restrict__ og)
{
    int qt = blockIdx.x, b = blockIdx.y;
    const float* Q = Qg + (size_t)b * L_ * DM_;
    const float* K = Kg + (size_t)b * L_ * DM_;
    const float* V = Vg + (size_t)b * L_ * DM_;

    __shared__ float Sred[16][16];
    __shared__ float P[16][16];
    __shared__ float rowscale[16];
    __shared__ float rowm[16], rowl[16];

    int tid  = threadIdx.x;
    int lane = tid & 31;
    int wv   = tid >> 5;
    int half = lane >> 4;
    int khalf = half << 1;
    int nlane = lane & 15;
    float invt = 1.0f / temp_p[0];

    if (tid < 16) { rowm[tid] = -3.0e38f; rowl[tid] = 0.f; }
    for (int i = tid; i < 256; i += 128) ((float*)Sred)[i] = 0.f;
    __syncthreads();

    int d0 = wv * 128;

    // preload this wave's Q A-fragments (16 rows x 128 cols -> 32 v2f frags)
    v2f qa[32];
    const float* qrow = Q + (size_t)(qt * 16 + nlane) * DM_ + d0 + khalf;
    #pragma unroll
    for (int s = 0; s < 32; ++s) qa[s] = *(const v2f*)(qrow + s * 4);

    v8f o[8];
    v8f z8 = {};
    #pragma unroll
    for (int i = 0; i < 8; ++i) o[i] = z8;

    for (int kt = 0; kt < L_ / 16; ++kt) {
        int k0 = kt * 16;
        // ---- S partial over this wave's 128 dims ----
        v8f c = {};
        const float* krow = K + (size_t)(k0 + nlane) * DM_ + d0 + khalf;
        #pragma unroll 8
        for (int s = 0; s < 32; ++s) {
            v2f bb = *(const v2f*)(krow + s * 4);
            c = wmma4(qa[s], bb, c);
        }
        #pragma unroll
        for (int r = 0; r < 8; ++r)
            atomicAdd(&Sred[r + half * 8][nlane], c[r]);
        __syncthreads();

        // ---- online softmax bookkeeping: one thread per q-row ----
        if (tid < 16) {
            int i = tid;
            float mloc = -3.0e38f;
            float sv[16];
            #pragma unroll
            for (int j = 0; j < 16; ++j) {
                sv[j] = Sred[i][j] * invt;
                mloc = fmaxf(mloc, sv[j]);
            }
            float mold = rowm[i];
            float mn = fmaxf(mold, mloc);
            float scl = __expf(mold - mn);
            float ssum = 0.f;
            #pragma unroll
            for (int j = 0; j < 16; ++j) {
                float e = __expf(sv[j] - mn);
                P[i][j] = e;
                ssum += e;
            }
            rowl[i] = rowl[i] * scl + ssum;
            rowm[i] = mn;
            rowscale[i] = scl;
            #pragma unroll
            for (int j = 0; j < 16; ++j) Sred[i][j] = 0.f;   // ready for next kt
        }
        __syncthreads();

        // ---- rescale O, then O += P @ V over this wave's 128 dims ----
        float rs[8];
        #pragma unroll
        for (int r = 0; r < 8; ++r) rs[r] = rowscale[r + half * 8];
        #pragma unroll
        for (int sub = 0; sub < 8; ++sub) {
            v8f oc = o[sub];
            #pragma unroll
            for (int r = 0; r < 8; ++r) oc[r] *= rs[r];
            int e0 = d0 + sub * 16;
            #pragma unroll
            for (int ksub = 0; ksub < 4; ++ksub) {
                v2f a = *(const v2f*)&P[nlane][ksub * 4 + khalf];   // ds_load_b64
                const float* vp = V + (size_t)(k0 + ksub * 4 + khalf) * DM_ + e0 + nlane;
                v2f bb; bb.x = vp[0]; bb.y = vp[DM_];
                oc = wmma4(a, bb, oc);
            }
            o[sub] = oc;
        }
        __syncthreads();
    }

    // ---- epilogue: divide by softmax denominator, write og ----
    float linv[8];
    #pragma unroll
    for (int r = 0; r < 8; ++r) linv[r] = 1.0f / rowl[r + half * 8];
    float* out = og + (size_t)b * L_ * DM_ + (size_t)qt * 16 * DM_;
    #pragma unroll
    for (int sub = 0; sub < 8; ++sub) {
        int e0 = d0 + sub * 16;
        #pragma unroll
        for (int r = 0; r < 8; ++r)
            out[(size_t)(r + half * 8) * DM_ + e0 + nlane] = o[sub][r] * linv[r];
    }
}

// ============================================================================
// Kernel 4: fuse (alpha blend, residual) + LayerNorm.  grid (L, B), block 128.
// Window count for the overlap-average is closed-form (lw == 1 identically).
// ============================================================================
__global__ void fuse_ln_kernel(
    const float* __restrict__ q, const float* __restrict__ acc,
    const float* __restrict__ og, const float* __restrict__ alpha_p,
    const float* __restrict__ gamma, const float* __restrict__ beta,
    float* __restrict__ z)
{
    int t = blockIdx.x, b = blockIdx.y;
    size_t base = ((size_t)b * L_ + t) * DM_;
    float alpha = alpha_p[0];

    int cnt = 0;
    #pragma unroll
    for (int w = 0; w < WIN_; ++w) {
        int n = t - w;
        if (n >= 0 && (n % STEP_) == 0 && (n / STEP_) < NW_) cnt++;
    }
    float inv = 1.0f / ((float)cnt + 1e-6f);

    __shared__ float red[128];
    int tid = threadIdx.x;
    float x[4];
    float lsum = 0.f;
    #pragma unroll
    for (int i = 0; i < 4; ++i) {
        int c = tid + i * 128;
        float outl = acc[base + c] * inv;
        float fu = alpha * outl + (1.f - alpha) * og[base + c];
        float xv = fu + q[base + c];
        x[i] = xv;
        lsum += xv;
    }
    red[tid] = lsum; __syncthreads();
    for (int s = 64; s > 0; s >>= 1) { if (tid < s) red[tid] += red[tid + s]; __syncthreads(); }
    float mean = red[0] / (float)DM_;
    __syncthreads();
    float lvar = 0.f;
    #pragma unroll
    for (int i = 0; i < 4; ++i) { float d = x[i] - mean; lvar += d * d; }
    red[tid] = lvar; __syncthreads();
    for (int s = 64; s > 0; s >>= 1) { if (tid < s) red[tid] += red[tid + s]; __syncthreads(); }
    float var = red[0] / (float)DM_;
    float rstd = rsqrtf(var + LN_EPS);
    #pragma unroll
    for (int i = 0; i < 4; ++i) {
        int c = tid + i * 128;
        z[base + c] = (x[i] - mean) * rstd * gamma[c] + beta[c];
    }
}

// ============================================================================
extern "C" void kernel_launch(void* const* d_in, const int* in_sizes, int n_in,
                              void* d_out, int out_size, void* d_ws, size_t ws_size,
                              hipStream_t stream)
{
    const float* q     = (const float*)d_in[0];
    const float* k     = (const float*)d_in[1];
    const float* v     = (const float*)d_in[2];
    const float* Wq_l  = (const float*)d_in[3];
    const float* Wk_l  = (const float*)d_in[4];
    const float* Wv_l  = (const float*)d_in[5];
    const float* Wq_g  = (const float*)d_in[6];
    const float* Wk_g  = (const float*)d_in[7];
    const float* Wv_g  = (const float*)d_in[8];
    const float* ltemp = (const float*)d_in[9];
    const float* gtemp = (const float*)d_in[10];
    const float* alpha = (const float*)d_in[11];
    const float* gamma = (const float*)d_in[12];
    const float* beta  = (const float*)d_in[13];

    const size_t NBLD = (size_t)B_ * L_ * DM_;     // 8,388,608 floats
    float* z_out  = (float*)d_out;
    float* aw_out = (float*)d_out + NBLD;

    float* acc = (float*)d_ws;                     // local-path accumulator
    float* og  = acc + NBLD;                       // global-attention output
    float* Qg  = og  + NBLD;
    float* Kg  = Qg  + NBLD;
    float* Vg  = Kg  + NBLD;                       // ~168 MB total workspace

    hipMemsetAsync(acc, 0, NBLD * sizeof(float), stream);

    local_attn_kernel<<<B_ * H_ * NW_, 128, 0, stream>>>(
        q, k, v, Wq_l, Wk_l, Wv_l, ltemp, acc, aw_out);

    proj_g_kernel<<<dim3(L_ / 16, B_, 3), 128, 0, stream>>>(
        q, k, v, Wq_g, Wk_g, Wv_g, Qg, Kg, Vg);

    global_attn_kernel<<<dim3(L_ / 16, B_), 128, 0, stream>>>(
        Qg, Kg, Vg, gtemp, og);

    fuse_ln_kernel<<<dim3(L_, B_), 128, 0, stream>>>(
        q, acc, og, alpha, gamma, beta, z_out);
}